// SimpleGNNLayer_71708773974823
// MI455X (gfx1250) — compile-verified
//
#include <hip/hip_runtime.h>

typedef __attribute__((ext_vector_type(2))) float v2f;
typedef __attribute__((ext_vector_type(8))) float v8f;

#define N_NODES 100000
#define D 128
#define N_EDGES 1600000
#define TILES (N_NODES / 16)       // 6250 exactly
#define WAVES_PER_BLOCK 8
#define LDS_ROW 132                 // 128 + 4 pad -> conflict-free column reads

// ---------------------------------------------------------------------------
// Kernel 1: zero aggregate buffer + counts (contiguous in workspace)
// ---------------------------------------------------------------------------
__global__ void gnn_zero_ws(float* __restrict__ p, int n) {
    int i = blockIdx.x * blockDim.x + threadIdx.x;
    int stride = gridDim.x * blockDim.x;
    for (; i < n; i += stride) p[i] = 0.0f;
}

// ---------------------------------------------------------------------------
// Kernel 2: edge scatter. One wave per edge (grid-stride).
// Lane l handles features [4l, 4l+3]: 32 lanes * 4 = 128 features.
// ---------------------------------------------------------------------------
__global__ void gnn_scatter(const float* __restrict__ x,
                            const int*   __restrict__ edges,
                            float*       __restrict__ agg,
                            float*       __restrict__ counts) {
    const int lane   = threadIdx.x & 31;
    const int wave   = (blockIdx.x * blockDim.x + threadIdx.x) >> 5;
    const int nwaves = (gridDim.x * blockDim.x) >> 5;

    for (int e = wave; e < N_EDGES; e += nwaves) {
        const int src = edges[e];             // row 0 of (2, N_EDGES)
        const int tgt = edges[N_EDGES + e];   // row 1

        const float4 xv = *(const float4*)(x + (size_t)src * D + lane * 4);
        float* dst = agg + (size_t)tgt * D + lane * 4;
        atomicAdd(dst + 0, xv.x);
        atomicAdd(dst + 1, xv.y);
        atomicAdd(dst + 2, xv.z);
        atomicAdd(dst + 3, xv.w);
        if (lane == 0) atomicAdd(counts + tgt, 1.0f);
    }
}

// ---------------------------------------------------------------------------
// Kernel 3: combined = x + agg/max(count,1); out = relu(combined @ W^T + b)
// One wave per 16-row tile; full f32 WMMA (16x16x4) so results stay exact.
// ---------------------------------------------------------------------------
__global__ void __launch_bounds__(WAVES_PER_BLOCK * 32)
gnn_gemm(const float* __restrict__ x,
         const float* __restrict__ W,
         const float* __restrict__ bias,
         const float* __restrict__ agg,
         const float* __restrict__ counts,
         float*       __restrict__ out) {
    __shared__ float lds[WAVES_PER_BLOCK][16][LDS_ROW];

    const int lane = threadIdx.x & 31;
    const int w    = threadIdx.x >> 5;                 // wave in block
    const int tile = blockIdx.x * WAVES_PER_BLOCK + w; // 16-row tile
    if (tile >= TILES) return;                         // wave-uniform: EXEC stays full
    const int row0 = tile * 16;

    // Stage combined[16][128] into LDS (each wave owns its region; no barrier
    // needed — intra-wave DScnt dependencies are handled by the compiler).
#pragma unroll
    for (int r = 0; r < 16; ++r) {
        const int row = row0 + r;
        const float inv = 1.0f / fmaxf(counts[row], 1.0f);
        const float4 xv = *(const float4*)(x   + (size_t)row * D + lane * 4);
        const float4 av = *(const float4*)(agg + (size_t)row * D + lane * 4);
        float4 cv;
        cv.x = xv.x + av.x * inv;
        cv.y = xv.y + av.y * inv;
        cv.z = xv.z + av.z * inv;
        cv.w = xv.w + av.w * inv;
        *(float4*)(&lds[w][r][lane * 4]) = cv;
    }

    // Fragment coordinates (ISA §7.12.2, 32-bit A 16x4 layout, B mirrored):
    const int m    = lane & 15;              // A: M row / B: N col / D: N col
    const int koff = (lane >> 4) << 1;       // lanes 0-15 -> K{0,1}; 16-31 -> K{2,3}

    v8f acc[8];
#pragma unroll
    for (int nt = 0; nt < 8; ++nt) acc[nt] = (v8f){0.f,0.f,0.f,0.f,0.f,0.f,0.f,0.f};

#pragma unroll 4
    for (int k0 = 0; k0 < D; k0 += 4) {
        v2f a;
        a.x = lds[w][m][k0 + koff + 0];
        a.y = lds[w][m][k0 + koff + 1];
#pragma unroll
        for (int nt = 0; nt < 8; ++nt) {
            const int n = nt * 16 + m;       // B[k][n] = W[n][k]  (out = combined @ W^T)
            v2f bf;
            bf.x = W[(size_t)n * D + k0 + koff + 0];
            bf.y = W[(size_t)n * D + k0 + koff + 1];
            acc[nt] = __builtin_amdgcn_wmma_f32_16x16x4_f32(
                false, a, false, bf, (short)0, acc[nt], false, false);
        }
    }

    // Epilogue: D layout — lane 0-15: N=lane, M=v; lane 16-31: N=lane-16, M=v+8
    const int mhi = (lane >> 4) << 3;
#pragma unroll
    for (int nt = 0; nt < 8; ++nt) {
        const float bv = bias[nt * 16 + m];
#pragma unroll
        for (int v = 0; v < 8; ++v) {
            float r = acc[nt][v] + bv;
            r = fmaxf(r, 0.0f);
            out[(size_t)(row0 + v + mhi) * D + nt * 16 + m] = r;
        }
    }
}

// ---------------------------------------------------------------------------
extern "C" void kernel_launch(void* const* d_in, const int* in_sizes, int n_in,
                              void* d_out, int out_size, void* d_ws, size_t ws_size,
                              hipStream_t stream) {
    const float* x     = (const float*)d_in[0];
    const int*   edges = (const int*)  d_in[1];
    const float* W     = (const float*)d_in[2];
    const float* b     = (const float*)d_in[3];
    float*       out   = (float*)d_out;

    float* agg    = (float*)d_ws;                       // 100000*128 floats
    float* counts = agg + (size_t)N_NODES * D;          // 100000 floats (contiguous)

    const int zero_n = N_NODES * D + N_NODES;
    gnn_zero_ws<<<2048, 256, 0, stream>>>(agg, zero_n);

    gnn_scatter<<<4096, 256, 0, stream>>>(x, edges, agg, counts);

    const int gemm_blocks = (TILES + WAVES_PER_BLOCK - 1) / WAVES_PER_BLOCK; // 782
    gnn_gemm<<<gemm_blocks, WAVES_PER_BLOCK * 32, 0, stream>>>(x, W, b, agg, counts, out);
}